// GNN_12481174962937
// MI455X (gfx1250) — compile-verified
//
#include <hip/hip_runtime.h>
#include <hip/hip_bf16.h>

typedef float v2f __attribute__((ext_vector_type(2)));
typedef float v8f __attribute__((ext_vector_type(8)));

#define N_NODES 50000
#define N_EDGES 800000
#define IN_CH   128
#define HID_CH  128
#define OUT_CH  64
#define N_GRAPHS 256

// Truncate a generic pointer to the 32-bit LDS offset (generic LDS address is
// {SHARED_BASE[63:32], lds_addr[31:0]} per the CDNA5 aperture mapping).
__device__ __forceinline__ unsigned lds_addr32(const void* p) {
    return (unsigned)(uintptr_t)p;
}

// Async DMA: copy 32 bytes (global -> LDS) per calling thread, no VGPR round
// trip; tracked by ASYNCcnt. INST_OFFSET adds to both LDS and global address
// (CDNA5 ISA), so offset:16 continues the same chunk.
__device__ __forceinline__ void async_copy32(unsigned lds, const void* g) {
    asm volatile(
        "global_load_async_to_lds_b128 %0, %1, off\n\t"
        "global_load_async_to_lds_b128 %0, %1, off offset:16"
        :
        : "v"(lds), "v"(g)
        : "memory");
}

__device__ __forceinline__ void async_wait_all() {
    asm volatile("s_wait_asynccnt 0x0" ::: "memory");
}

// ---------------------------------------------------------------------------
// 0) init: zero agg + pooled, deg := 1.0 (self loop)
// ---------------------------------------------------------------------------
__global__ void init_kernel(float* __restrict__ agg, float* __restrict__ pooled,
                            float* __restrict__ dinv) {
    int i = blockIdx.x * blockDim.x + threadIdx.x;           // exactly 6.4M threads
    agg[i] = 0.0f;
    if (i < N_GRAPHS * HID_CH) pooled[i] = 0.0f;
    if (i < N_NODES)           dinv[i]   = 1.0f;             // deg starts at 1 (self loop)
}

// ---------------------------------------------------------------------------
// 1) degree accumulation over edge destinations
// ---------------------------------------------------------------------------
__global__ void deg_kernel(const int* __restrict__ dst, float* __restrict__ deg) {
    int e = blockIdx.x * blockDim.x + threadIdx.x;
    if (e < N_EDGES) atomicAdd(&deg[dst[e]], 1.0f);
}

// ---------------------------------------------------------------------------
// 2) dinv = rsqrt(deg), in place
// ---------------------------------------------------------------------------
__global__ void rsqrt_kernel(float* __restrict__ dinv) {
    int n = blockIdx.x * blockDim.x + threadIdx.x;
    if (n < N_NODES) dinv[n] = __frsqrt_rn(dinv[n]);
}

// ---------------------------------------------------------------------------
// 3) h = x @ W1   (50000x128 @ 128x128) via V_WMMA_F32_16X16X4_F32
//    block = 256 threads = 8 waves; block owns one 16-row M strip,
//    wave w owns N-tile w (16 cols). x strip staged in LDS via async
//    global->LDS DMA; W1 streamed from L2 (massive reuse, 64KB).
// ---------------------------------------------------------------------------
__global__ void gemm1_wmma(const float* __restrict__ x, const float* __restrict__ W1,
                           float* __restrict__ h) {
    __shared__ float xs[16 * IN_CH];                         // 8 KB
    const int tid   = threadIdx.x;
    const int wave  = tid >> 5;
    const int lane  = tid & 31;
    const int mtile = blockIdx.x;                            // 3125 tiles, exact

    // stage 16x128 strip of x: 2048 floats; 256 threads x 32B async DMA
    {
        const float* xg = x + (size_t)mtile * (16 * IN_CH) + tid * 8;
        async_copy32(lds_addr32(xs) + tid * 32u, xg);
        async_wait_all();
    }
    __syncthreads();

    const int row = lane & 15;             // A: M = lane%16
    const int kb  = (lane >> 4) << 1;      // A/B: lanes 0-15 -> K {0,1}, lanes 16-31 -> K {2,3}
    const int col = (wave << 4) + (lane & 15);               // B/D: N within 0..127

    v8f acc = {};
    #pragma unroll
    for (int k = 0; k < IN_CH; k += 4) {
        v2f a = { xs[row * IN_CH + k + kb], xs[row * IN_CH + k + kb + 1] };
        v2f b = { W1[(k + kb) * HID_CH + col], W1[(k + kb + 1) * HID_CH + col] };
        acc = __builtin_amdgcn_wmma_f32_16x16x4_f32(false, a, false, b,
                                                    (short)0, acc, false, false);
    }

    // D layout: VGPR i -> M=i (lanes 0-15) / M=8+i (lanes 16-31), N = lane%16
    const int mbase = mtile * 16 + ((lane >> 4) << 3);
    #pragma unroll
    for (int i = 0; i < 8; ++i)
        h[(size_t)(mbase + i) * HID_CH + col] = acc[i];
}

// ---------------------------------------------------------------------------
// 4) edge scatter: agg[dst] += h[src] * dinv[src]*dinv[dst]
//    one wave per edge, float4 per lane (32*4 = 128 ch), f32 global atomics
// ---------------------------------------------------------------------------
__global__ void scatter_kernel(const int* __restrict__ src, const int* __restrict__ dst,
                               const float* __restrict__ dinv,
                               const float* __restrict__ h, float* __restrict__ agg) {
    const int e    = blockIdx.x * 8 + (threadIdx.x >> 5);    // 100000 blocks * 8 = 800000 exact
    const int lane = threadIdx.x & 31;
    const int q    = lane << 2;

    const int s = src[e];
    const int d = dst[e];
    const float nrm = dinv[s] * dinv[d];

    float4 hv = *(const float4*)&h[(size_t)s * HID_CH + q];
    float* ap = &agg[(size_t)d * HID_CH + q];
    atomicAdd(ap + 0, hv.x * nrm);
    atomicAdd(ap + 1, hv.y * nrm);
    atomicAdd(ap + 2, hv.z * nrm);
    atomicAdd(ap + 3, hv.w * nrm);
}

// ---------------------------------------------------------------------------
// 5) self loop + bias + ReLU + global max pool (bitwise atomicMax is exact
//    for the non-negative post-ReLU values; pooled pre-zeroed -> empty
//    graphs end up 0, matching where(isfinite, ., 0))
// ---------------------------------------------------------------------------
__global__ void pool_kernel(const float* __restrict__ h, const float* __restrict__ agg,
                            const float* __restrict__ dinv, const float* __restrict__ b1,
                            const int* __restrict__ batch, float* __restrict__ pooled) {
    const int g = blockIdx.x * blockDim.x + threadIdx.x;     // 50000*32 exact
    const int node = g >> 5;
    const int q    = (g & 31) << 2;

    const float di = dinv[node];
    const float sn = di * di;                                // self-loop norm
    const int   bg = batch[node];

    float4 hv = *(const float4*)&h[(size_t)node * HID_CH + q];
    float4 av = *(const float4*)&agg[(size_t)node * HID_CH + q];

    int* pp = (int*)&pooled[(size_t)bg * HID_CH + q];
    float v;
    v = fmaxf(av.x + hv.x * sn + b1[q + 0], 0.0f); atomicMax(pp + 0, __float_as_int(v));
    v = fmaxf(av.y + hv.y * sn + b1[q + 1], 0.0f); atomicMax(pp + 1, __float_as_int(v));
    v = fmaxf(av.z + hv.z * sn + b1[q + 2], 0.0f); atomicMax(pp + 2, __float_as_int(v));
    v = fmaxf(av.w + hv.w * sn + b1[q + 3], 0.0f); atomicMax(pp + 3, __float_as_int(v));
}

// ---------------------------------------------------------------------------
// 6) out = relu(pooled @ W2 + b2)  (256x128 @ 128x64) via WMMA f32
//    16 blocks x 128 threads (4 waves); wave w owns N-tile w.
// ---------------------------------------------------------------------------
__global__ void gemm2_wmma(const float* __restrict__ pooled, const float* __restrict__ W2,
                           const float* __restrict__ b2, float* __restrict__ out) {
    __shared__ float ps[16 * HID_CH];                        // 8 KB
    const int tid   = threadIdx.x;
    const int wave  = tid >> 5;
    const int lane  = tid & 31;
    const int mtile = blockIdx.x;                            // 16 tiles, exact

    {
        const float* pg = pooled + (size_t)mtile * (16 * HID_CH) + tid * 16;
        unsigned lds = lds_addr32(ps) + tid * 64u;           // 64 B per thread
        async_copy32(lds,       pg);
        async_copy32(lds + 32u, pg + 8);
        async_wait_all();
    }
    __syncthreads();

    const int row = lane & 15;
    const int kb  = (lane >> 4) << 1;
    const int col = (wave << 4) + (lane & 15);               // 0..63

    v8f acc = {};
    #pragma unroll
    for (int k = 0; k < HID_CH; k += 4) {
        v2f a = { ps[row * HID_CH + k + kb], ps[row * HID_CH + k + kb + 1] };
        v2f b = { W2[(k + kb) * OUT_CH + col], W2[(k + kb + 1) * OUT_CH + col] };
        acc = __builtin_amdgcn_wmma_f32_16x16x4_f32(false, a, false, b,
                                                    (short)0, acc, false, false);
    }

    const float bias = b2[col];
    const int mbase = mtile * 16 + ((lane >> 4) << 3);
    #pragma unroll
    for (int i = 0; i < 8; ++i)
        out[(size_t)(mbase + i) * OUT_CH + col] = fmaxf(acc[i] + bias, 0.0f);
}

// ---------------------------------------------------------------------------
extern "C" void kernel_launch(void* const* d_in, const int* in_sizes, int n_in,
                              void* d_out, int out_size, void* d_ws, size_t ws_size,
                              hipStream_t stream) {
    const float* x     = (const float*)d_in[0];
    const int*   ei    = (const int*)d_in[1];                // [2, E] flat
    const int*   batch = (const int*)d_in[2];
    const float* W1    = (const float*)d_in[3];
    const float* b1    = (const float*)d_in[4];
    const float* W2    = (const float*)d_in[5];
    const float* b2    = (const float*)d_in[6];
    float*       out   = (float*)d_out;

    const int* src = ei;                                     // edge_index[0]
    const int* dst = ei + N_EDGES;                           // edge_index[1]

    float* ws     = (float*)d_ws;
    float* h      = ws;                                      // 6,400,000
    float* agg    = ws + (size_t)N_NODES * HID_CH;           // 6,400,000
    float* dinv   = agg + (size_t)N_NODES * HID_CH;          // 50,000
    float* pooled = dinv + N_NODES;                          // 32,768

    // 0) init (6.4M threads exactly covers agg; pooled/deg covered by sub-ranges)
    init_kernel<<<(N_NODES * HID_CH) / 256, 256, 0, stream>>>(agg, pooled, dinv);
    // 1) degrees
    deg_kernel<<<(N_EDGES + 255) / 256, 256, 0, stream>>>(dst, dinv);
    // 2) rsqrt
    rsqrt_kernel<<<(N_NODES + 255) / 256, 256, 0, stream>>>(dinv);
    // 3) h = x @ W1
    gemm1_wmma<<<N_NODES / 16, 256, 0, stream>>>(x, W1, h);
    // 4) edge scatter
    scatter_kernel<<<N_EDGES / 8, 256, 0, stream>>>(src, dst, dinv, h, agg);
    // 5) self loop + bias + relu + max pool
    pool_kernel<<<(N_NODES * 32) / 256, 256, 0, stream>>>(h, agg, dinv, b1, batch, pooled);
    // 6) out = relu(pooled @ W2 + b2)
    gemm2_wmma<<<N_GRAPHS / 16, 128, 0, stream>>>(pooled, W2, b2, out);
}